// ScalarPotentialLM_46540265620098
// MI455X (gfx1250) — compile-verified
//
#include <hip/hip_runtime.h>
#include <math.h>

#define D     128
#define T_LEN 512
#define BATCH 4
#define HID   512
#define VOCAB 50257
#define NPOS  (BATCH * T_LEN)
#define LROW  516          // padded LDS row stride (floats): 516 % 64 = 4 -> conflict-free
#define AROW  132          // padded A-tile row stride for logits kernel

typedef float v2f __attribute__((ext_vector_type(2)));
typedef float v8f __attribute__((ext_vector_type(8)));

__device__ __forceinline__ v2f ldv2(const float* p) {        // 8B-aligned b64 load
  return *(const v2f*)p;
}

// D(16x16,f32) = A(16x4,f32) * B(4x16,f32) + C  -- full-precision CDNA5 WMMA
__device__ __forceinline__ v8f wmma4(v2f a, v2f b, v8f c) {
  return __builtin_amdgcn_wmma_f32_16x16x4_f32(false, a, false, b, (short)0, c,
                                               false, false);
}

__device__ __forceinline__ v8f zero8() {
  v8f z;
#pragma unroll
  for (int i = 0; i < 8; ++i) z[i] = 0.0f;
  return z;
}

__device__ __forceinline__ float gelu_f(float x) {
  return 0.5f * x * (1.0f + erff(x * 0.7071067811865476f));
}
__device__ __forceinline__ float gelu_prime_f(float x) {
  float cdf = 0.5f * (1.0f + erff(x * 0.7071067811865476f));
  float pdf = 0.3989422804014327f * expf(-0.5f * x * x);
  return fmaf(x, pdf, cdf);
}

// ---------------------------------------------------------------------------
// emb[p,d] = E[x[p],d] + P[t,d]; h = emb; v = 0
__global__ void embed_kernel(const int* __restrict__ x, const float* __restrict__ E,
                             const float* __restrict__ P, float* __restrict__ emb,
                             float* __restrict__ h, float* __restrict__ v) {
  int p = blockIdx.x;            // 0..NPOS-1
  int d = threadIdx.x;           // 0..127
  int t = p & (T_LEN - 1);
  int tok = x[p];
  float e = E[(size_t)tok * D + d] + P[(size_t)t * D + d];
  size_t idx = (size_t)p * D + d;
  emb[idx] = e;
  h[idx]   = e;
  v[idx]   = 0.0f;
}

// xi[b,t,d] = cumsum_t(emb)/(t+1); one thread per (b,d) channel
__global__ void cumsum_kernel(const float* __restrict__ emb, float* __restrict__ xi) {
  int b = blockIdx.x;
  int d = threadIdx.x;
  float acc = 0.0f;
  for (int t = 0; t < T_LEN; ++t) {
    size_t idx = ((size_t)b * T_LEN + t) * D + d;
    acc += emb[idx];
    xi[idx] = acc / (float)(t + 1);
  }
}

// WT[n*K + k] = W[k*N + n]; LDS-tiled 32x32; K,N multiples of 32
__global__ __launch_bounds__(256) void transpose_kernel(const float* __restrict__ W,
                                                        float* __restrict__ WT,
                                                        int K, int N) {
  __shared__ float tile[32][33];
  const int kb = blockIdx.x * 32;
  const int nb = blockIdx.y * 32;
  const int tx = threadIdx.x;        // 0..31
  const int ty = threadIdx.y;        // 0..7
#pragma unroll
  for (int r = ty; r < 32; r += 8)
    tile[r][tx] = W[(size_t)(kb + r) * N + (nb + tx)];
  __syncthreads();
#pragma unroll
  for (int r = ty; r < 32; r += 8)
    WT[(size_t)(nb + r) * K + (kb + tx)] = tile[tx][r];
}

// ---------------------------------------------------------------------------
// One dynamics step for a 16-position tile: MLP forward + backward wrt h,
// then v = (v - grad/m)/(1+gamma), h += v.   (DT == 1)
// Block = 128 threads = 4 waves; wave w owns output columns [w*128, w*128+128).
// Forward B-operands come from the pre-transposed weights (contiguous b64);
// backward B-operands read the original weights (already [n][k]-contiguous).
// All GELU / GELU' values are computed exactly once at activation-store time.
// LDS buffer aliasing:
//   bufA: g1 (fwd L2 input)  -> e3 = W3*gelu'(a3) (bwd3 A) -> da2 (bwd2 A)
//   bufB: g2 (fwd L3 input)  -> da1 (bwd1 A)
//   bufC: d1 = gelu'(a1)
//   bufD: d2 = gelu'(a2)
__global__ __launch_bounds__(128) void mlp_step_kernel(
    const float* __restrict__ xi, float* __restrict__ h, float* __restrict__ v,
    const float* __restrict__ W0, const float* __restrict__ W0T,
    const float* __restrict__ b0,
    const float* __restrict__ W1, const float* __restrict__ W1T,
    const float* __restrict__ b1,
    const float* __restrict__ W2, const float* __restrict__ W2T,
    const float* __restrict__ b2,
    const float* __restrict__ W3,
    const float* __restrict__ raw_m, const float* __restrict__ raw_g) {
  __shared__ float bufA[16 * LROW];
  __shared__ float bufB[16 * LROW];
  __shared__ float bufC[16 * LROW];
  __shared__ float bufD[16 * LROW];

  const int lane = threadIdx.x & 31;
  const int wave = threadIdx.x >> 5;
  const int hf   = lane >> 4;    // lane half (K split for A/B operands)
  const int lm   = lane & 15;    // row (A) / column (B,C,D) within tile
  const int pos0 = blockIdx.x * 16;
  const int n0w  = wave * 128;

  v8f acc[8];

  // ---------------- forward layer 1: a1 = [xi,h] @ W0 + b0  (K = 256) -------
#pragma unroll
  for (int nt = 0; nt < 8; ++nt) acc[nt] = zero8();
  const size_t rowA = (size_t)(pos0 + lm) * D;
  for (int k0 = 0; k0 < D; k0 += 4) {          // first half: z0 = xi
    int ka = k0 + 2 * hf;
    v2f a = ldv2(xi + rowA + ka);
#pragma unroll
    for (int nt = 0; nt < 8; ++nt) {
      int n = n0w + nt * 16 + lm;
      v2f b = ldv2(W0T + (size_t)n * (2 * D) + ka);      // = {W0[ka][n], W0[ka+1][n]}
      acc[nt] = wmma4(a, b, acc[nt]);
    }
  }
  for (int k0 = 0; k0 < D; k0 += 4) {          // second half: z0 = h
    int ka = k0 + 2 * hf;
    v2f a = ldv2(h + rowA + ka);
#pragma unroll
    for (int nt = 0; nt < 8; ++nt) {
      int n = n0w + nt * 16 + lm;
      v2f b = ldv2(W0T + (size_t)n * (2 * D) + D + ka);
      acc[nt] = wmma4(a, b, acc[nt]);
    }
  }
#pragma unroll
  for (int nt = 0; nt < 8; ++nt) {
    int n = n0w + nt * 16 + lm;
    float bias = b0[n];
#pragma unroll
    for (int j = 0; j < 8; ++j) {
      float aval = acc[nt][j] + bias;
      int mm = j + 8 * hf;
      bufA[mm * LROW + n] = gelu_f(aval);        // g1
      bufC[mm * LROW + n] = gelu_prime_f(aval);  // d1
    }
  }
  __syncthreads();

  // ---------------- forward layer 2: a2 = g1 @ W1 + b1  (K = 512) -----------
#pragma unroll
  for (int nt = 0; nt < 8; ++nt) acc[nt] = zero8();
  for (int k0 = 0; k0 < HID; k0 += 4) {
    int ka = k0 + 2 * hf;
    v2f a = ldv2(&bufA[lm * LROW + ka]);
#pragma unroll
    for (int nt = 0; nt < 8; ++nt) {
      int n = n0w + nt * 16 + lm;
      v2f b = ldv2(W1T + (size_t)n * HID + ka);
      acc[nt] = wmma4(a, b, acc[nt]);
    }
  }
#pragma unroll
  for (int nt = 0; nt < 8; ++nt) {
    int n = n0w + nt * 16 + lm;
    float bias = b1[n];
#pragma unroll
    for (int j = 0; j < 8; ++j) {
      float aval = acc[nt][j] + bias;
      int mm = j + 8 * hf;
      bufB[mm * LROW + n] = gelu_f(aval);        // g2
      bufD[mm * LROW + n] = gelu_prime_f(aval);  // d2
    }
  }
  __syncthreads();

  // ------- forward layer 3: a3 = g2 @ W2 + b2 ; store e3 = W3*gelu'(a3) -----
#pragma unroll
  for (int nt = 0; nt < 8; ++nt) acc[nt] = zero8();
  for (int k0 = 0; k0 < HID; k0 += 4) {
    int ka = k0 + 2 * hf;
    v2f a = ldv2(&bufB[lm * LROW + ka]);
#pragma unroll
    for (int nt = 0; nt < 8; ++nt) {
      int n = n0w + nt * 16 + lm;
      v2f b = ldv2(W2T + (size_t)n * HID + ka);
      acc[nt] = wmma4(a, b, acc[nt]);
    }
  }
#pragma unroll
  for (int nt = 0; nt < 8; ++nt) {
    int n = n0w + nt * 16 + lm;
    float bias = b2[n];
    float w3n  = W3[n];
#pragma unroll
    for (int j = 0; j < 8; ++j) {
      float aval = acc[nt][j] + bias;
      int mm = j + 8 * hf;
      bufA[mm * LROW + n] = w3n * gelu_prime_f(aval);  // e3 (bwd3 A-matrix)
    }
  }
  __syncthreads();

  // ---------------- backward 3: dg2 = e3 @ W2^T  ----------------------------
#pragma unroll
  for (int nt = 0; nt < 8; ++nt) acc[nt] = zero8();
  for (int k0 = 0; k0 < HID; k0 += 4) {
    int ka = k0 + 2 * hf;
    v2f a = ldv2(&bufA[lm * LROW + ka]);
#pragma unroll
    for (int nt = 0; nt < 8; ++nt) {
      int n = n0w + nt * 16 + lm;
      v2f b = ldv2(W2 + (size_t)n * HID + ka);   // B[k][n] = W2[n][k]
      acc[nt] = wmma4(a, b, acc[nt]);
    }
  }
  __syncthreads();                             // everyone done reading e3
#pragma unroll
  for (int nt = 0; nt < 8; ++nt) {             // da2 = dg2 * d2 -> bufA
    int n = n0w + nt * 16 + lm;
#pragma unroll
    for (int j = 0; j < 8; ++j) {
      int mm = j + 8 * hf;
      bufA[mm * LROW + n] = acc[nt][j] * bufD[mm * LROW + n];
    }
  }
  __syncthreads();

  // ---------------- backward 2: dg1 = da2 @ W1^T  ---------------------------
#pragma unroll
  for (int nt = 0; nt < 8; ++nt) acc[nt] = zero8();
  for (int k0 = 0; k0 < HID; k0 += 4) {
    int ka = k0 + 2 * hf;
    v2f a = ldv2(&bufA[lm * LROW + ka]);
#pragma unroll
    for (int nt = 0; nt < 8; ++nt) {
      int n = n0w + nt * 16 + lm;
      v2f b = ldv2(W1 + (size_t)n * HID + ka);   // B[k][n] = W1[n][k]
      acc[nt] = wmma4(a, b, acc[nt]);
    }
  }
  __syncthreads();                             // everyone done reading da2
#pragma unroll
  for (int nt = 0; nt < 8; ++nt) {             // da1 = dg1 * d1 -> bufB
    int n = n0w + nt * 16 + lm;
#pragma unroll
    for (int j = 0; j < 8; ++j) {
      int mm = j + 8 * hf;
      bufB[mm * LROW + n] = acc[nt][j] * bufC[mm * LROW + n];
    }
  }
  __syncthreads();

  // -------- backward 1 (h part): grad = da1 @ W0[128:,:]^T   (16 x 128) -----
  v8f gacc[2];
  gacc[0] = zero8();
  gacc[1] = zero8();
  for (int k0 = 0; k0 < HID; k0 += 4) {
    int ka = k0 + 2 * hf;
    v2f a = ldv2(&bufB[lm * LROW + ka]);
#pragma unroll
    for (int nt = 0; nt < 2; ++nt) {
      int n = wave * 32 + nt * 16 + lm;
      v2f b = ldv2(W0 + (size_t)(D + n) * HID + ka);  // B[k][n] = W0[128+n][k]
      gacc[nt] = wmma4(a, b, gacc[nt]);
    }
  }

  // -------- update v, h  (DT = 1): v = (v - g/m)/(1+gamma); h += v ----------
  float ms  = log1pf(expf(raw_m[0])) + 0.001f;   // softplus + 0.001
  float ga  = log1pf(expf(raw_g[0]));            // softplus
  float rms = 1.0f / ms;
  float inv = 1.0f / (1.0f + ga);
#pragma unroll
  for (int nt = 0; nt < 2; ++nt) {
    int n = wave * 32 + nt * 16 + lm;
#pragma unroll
    for (int j = 0; j < 8; ++j) {
      int mm = j + 8 * hf;
      size_t idx = (size_t)(pos0 + mm) * D + n;
      float vn = (v[idx] - gacc[nt][j] * rms) * inv;
      v[idx] = vn;
      h[idx] = h[idx] + vn;
    }
  }
}

// ---------------------------------------------------------------------------
// logits = h @ E^T : 16x16 WMMA tiles, K = 128. Output-write bound (412 MB).
// A-tile (16x128 of h) staged once per block in padded LDS; each of the 4
// waves computes 2 N-tiles (block covers 8 vocab tiles).
__global__ __launch_bounds__(128) void logits_kernel(const float* __restrict__ h,
                                                     const float* __restrict__ E,
                                                     float* __restrict__ out) {
  __shared__ float sh_a[16 * AROW];

  const int tid  = threadIdx.x;
  const int lane = tid & 31;
  const int wave = tid >> 5;
  const int hf   = lane >> 4;
  const int lm   = lane & 15;
  const int m0   = blockIdx.y * 16;

  // cooperative stage of A tile: 16 rows x 128 cols, coalesced
#pragma unroll
  for (int r = 0; r < 16; ++r)
    sh_a[r * AROW + tid] = h[(size_t)(m0 + r) * D + tid];
  __syncthreads();

  v8f acc0 = zero8();
  v8f acc1 = zero8();
  const int n0_0 = (blockIdx.x * 8 + wave * 2 + 0) * 16;
  const int n0_1 = (blockIdx.x * 8 + wave * 2 + 1) * 16;
  int nc0 = n0_0 + lm; if (nc0 > VOCAB - 1) nc0 = VOCAB - 1;  // clamp tails:
  int nc1 = n0_1 + lm; if (nc1 > VOCAB - 1) nc1 = VOCAB - 1;  // EXEC stays all-1s
  const size_t rowB0 = (size_t)nc0 * D;
  const size_t rowB1 = (size_t)nc1 * D;

  for (int k0 = 0; k0 < D; k0 += 4) {
    int ka = k0 + 2 * hf;
    v2f a  = ldv2(&sh_a[lm * AROW + ka]);
    v2f b0 = ldv2(E + rowB0 + ka);               // B[k][n] = E[n][k]
    v2f b1 = ldv2(E + rowB1 + ka);
    acc0 = wmma4(a, b0, acc0);
    acc1 = wmma4(a, b1, acc1);
  }

  int n0 = n0_0 + lm;
  if (n0 < VOCAB) {
#pragma unroll
    for (int j = 0; j < 8; ++j)
      out[(size_t)(m0 + j + 8 * hf) * VOCAB + n0] = acc0[j];
  }
  int n1 = n0_1 + lm;
  if (n1 < VOCAB) {
#pragma unroll
    for (int j = 0; j < 8; ++j)
      out[(size_t)(m0 + j + 8 * hf) * VOCAB + n1] = acc1[j];
  }
}

// ---------------------------------------------------------------------------
extern "C" void kernel_launch(void* const* d_in, const int* in_sizes, int n_in,
                              void* d_out, int out_size, void* d_ws, size_t ws_size,
                              hipStream_t stream) {
  const int*   x     = (const int*)d_in[0];
  const float* E     = (const float*)d_in[1];
  const float* P     = (const float*)d_in[2];
  const float* W0    = (const float*)d_in[3];
  const float* b0    = (const float*)d_in[4];
  const float* W1    = (const float*)d_in[5];
  const float* b1    = (const float*)d_in[6];
  const float* W2    = (const float*)d_in[7];
  const float* b2    = (const float*)d_in[8];
  const float* W3    = (const float*)d_in[9];
  // d_in[10] = b3: does not affect grad or logits
  const float* raw_m = (const float*)d_in[11];
  const float* raw_g = (const float*)d_in[12];

  float* ws  = (float*)d_ws;
  float* emb = ws;                                  // NPOS*D floats
  float* xi  = ws + (size_t)NPOS * D;
  float* h   = ws + (size_t)2 * NPOS * D;
  float* v   = ws + (size_t)3 * NPOS * D;
  float* w0t = ws + (size_t)4 * NPOS * D;           // HID x 2D
  float* w1t = w0t + (size_t)HID * 2 * D;           // HID x HID
  float* w2t = w1t + (size_t)HID * HID;             // HID x HID
  float* out = (float*)d_out;

  embed_kernel<<<NPOS, D, 0, stream>>>(x, E, P, emb, h, v);
  cumsum_kernel<<<BATCH, D, 0, stream>>>(emb, xi);
  transpose_kernel<<<dim3((2 * D) / 32, HID / 32), dim3(32, 8), 0, stream>>>(W0, w0t, 2 * D, HID);
  transpose_kernel<<<dim3(HID / 32, HID / 32), dim3(32, 8), 0, stream>>>(W1, w1t, HID, HID);
  transpose_kernel<<<dim3(HID / 32, HID / 32), dim3(32, 8), 0, stream>>>(W2, w2t, HID, HID);

  for (int s = 0; s < 8; ++s) {
    mlp_step_kernel<<<NPOS / 16, 128, 0, stream>>>(xi, h, v,
                                                   W0, w0t, b0,
                                                   W1, w1t, b1,
                                                   W2, w2t, b2,
                                                   W3, raw_m, raw_g);
  }

  const int n_tiles  = (VOCAB + 15) / 16;            // 3142
  dim3 lgrid((n_tiles + 7) / 8, NPOS / 16);          // (393, 128)
  logits_kernel<<<lgrid, 128, 0, stream>>>(h, E, out);
}